// ResidualGINLayer_44555990728952
// MI455X (gfx1250) — compile-verified
//
#include <hip/hip_runtime.h>

#define N_NODES 50000
#define IN_DIM 128
#define HID 128
#define N_EDGES 600000
#define BN_EPS 1e-5f

typedef __attribute__((ext_vector_type(2))) float v2f;
typedef __attribute__((ext_vector_type(8))) float v8f;

// ---------------------------------------------------------------------------
// Kernel 1: h = x (copy, float4), and zero the 256 stats floats (sum, sumsq)
// ---------------------------------------------------------------------------
__global__ void init_h_kernel(const float* __restrict__ x, float* __restrict__ h,
                              float* __restrict__ stats) {
  int tid = blockIdx.x * blockDim.x + threadIdx.x;
  int stride = gridDim.x * blockDim.x;
  const int n4 = N_NODES * IN_DIM / 4;
  const float4* x4 = (const float4*)x;
  float4* h4 = (float4*)h;
  for (int i = tid; i < n4; i += stride) h4[i] = x4[i];
  if (tid < 256) stats[tid] = 0.0f;
}

// ---------------------------------------------------------------------------
// Kernel 2: GIN aggregation  h[dst] += x[src]  (one wave per edge,
// each lane handles 4 consecutive floats of the 128-float row)
// ---------------------------------------------------------------------------
__global__ void scatter_kernel(const float* __restrict__ x,
                               const long long* __restrict__ ei,
                               float* __restrict__ h) {
  unsigned tid = blockIdx.x * blockDim.x + threadIdx.x;
  unsigned e = tid >> 5;
  unsigned lane = tid & 31u;
  if (e >= N_EDGES) return;
  int src = (int)ei[e];
  int dst = (int)ei[N_EDGES + e];
  float4 v = *(const float4*)(x + (size_t)src * IN_DIM + lane * 4);
  float* p = h + (size_t)dst * IN_DIM + lane * 4;
  atomicAdd(p + 0, v.x);
  atomicAdd(p + 1, v.y);
  atomicAdd(p + 2, v.z);
  atomicAdd(p + 3, v.w);
}

// ---------------------------------------------------------------------------
// Kernel 3: h1 = h @ W1^T + b1 via V_WMMA_F32_16X16X4_F32.
// One wave = one 16x16 output tile; K=128 consumed in 32 WMMA steps.
// A layout: lane L holds row L%16, K offsets 2*(L/16)+{0,1} per step.
// B (K x N) mirrors A with N = lane%16 and B[k][n] = W1[colBase+n][k].
// ---------------------------------------------------------------------------
__global__ void gemm1_kernel(const float* __restrict__ h, const float* __restrict__ W1,
                             const float* __restrict__ b1, float* __restrict__ h1) {
  int wave = (blockIdx.x * blockDim.x + threadIdx.x) >> 5;
  int lane = threadIdx.x & 31;
  int tileRow = wave >> 3;            // HID/16 = 8 col tiles
  int tileCol = wave & 7;
  int rowBase = tileRow * 16;
  int colBase = tileCol * 16;
  int m = lane & 15;
  int kh = (lane >> 4) << 1;          // 0 or 2
  const float* arow = h  + (size_t)(rowBase + m) * IN_DIM + kh;
  const float* brow = W1 + (size_t)(colBase + m) * IN_DIM + kh;
  v8f c = {};
#pragma unroll
  for (int k0 = 0; k0 < IN_DIM; k0 += 4) {
    v2f a, b;
    a.x = arow[k0];  a.y = arow[k0 + 1];
    b.x = brow[k0];  b.y = brow[k0 + 1];
    c = __builtin_amdgcn_wmma_f32_16x16x4_f32(false, a, false, b, (short)0, c,
                                              false, false);
  }
  int col = colBase + m;
  float bias = b1[col];
  int r0 = rowBase + ((lane >> 4) << 3);   // C: VGPR i -> row i (+8 for hi lanes)
#pragma unroll
  for (int i = 0; i < 8; ++i)
    h1[(size_t)(r0 + i) * HID + col] = c[i] + bias;
}

// ---------------------------------------------------------------------------
// Kernel 4: per-column sum / sum-of-squares of h1 (for batch-norm stats).
// Thread t accumulates column t%128 over strided rows; LDS-reduce the two
// half-blocks, then one atomicAdd per column per block.
// ---------------------------------------------------------------------------
__global__ void stats_kernel(const float* __restrict__ h1, float* __restrict__ sum,
                             float* __restrict__ sumsq) {
  const int rowsPerBlock = (N_NODES + gridDim.x - 1) / gridDim.x;
  int r0 = blockIdx.x * rowsPerBlock + (threadIdx.x >> 7);
  int r1 = (int)((blockIdx.x + 1) * rowsPerBlock);
  if (r1 > N_NODES) r1 = N_NODES;
  int col = threadIdx.x & 127;
  float s = 0.f, s2 = 0.f;
  for (int r = r0; r < r1; r += 2) {
    float v = h1[(size_t)r * HID + col];
    s += v;
    s2 += v * v;
  }
  __shared__ float ls[256];
  __shared__ float ls2[256];
  ls[threadIdx.x] = s;
  ls2[threadIdx.x] = s2;
  __syncthreads();
  if (threadIdx.x < 128) {
    atomicAdd(&sum[col],   ls[threadIdx.x]  + ls[threadIdx.x + 128]);
    atomicAdd(&sumsq[col], ls2[threadIdx.x] + ls2[threadIdx.x + 128]);
  }
}

// ---------------------------------------------------------------------------
// Kernel 5: fold BN into affine scale/shift:
//   scale = gamma * rsqrt(var + eps),  shift = beta - mu * scale
// ---------------------------------------------------------------------------
__global__ void finalize_kernel(const float* __restrict__ sum,
                                const float* __restrict__ sumsq,
                                const float* __restrict__ gamma,
                                const float* __restrict__ beta,
                                float* __restrict__ scale,
                                float* __restrict__ shift) {
  int c = threadIdx.x;
  if (c >= HID) return;
  const float invN = 1.0f / (float)N_NODES;
  float mu = sum[c] * invN;
  float var = sumsq[c] * invN - mu * mu;
  float rs = rsqrtf(var + BN_EPS);
  float sc = gamma[c] * rs;
  scale[c] = sc;
  shift[c] = beta[c] - mu * sc;
}

// ---------------------------------------------------------------------------
// Kernel 6: out = leaky( leaky(BN(h1),0.01) @ W2^T + b2 + x @ Wres^T, 0.2 )
// Both GEMMs accumulate into the same WMMA accumulator. BN + LeakyReLU(0.01)
// applied on-the-fly to the A operand (scale/shift cached in LDS).
// ---------------------------------------------------------------------------
__global__ void gemm2_kernel(const float* __restrict__ h1, const float* __restrict__ x,
                             const float* __restrict__ W2, const float* __restrict__ Wres,
                             const float* __restrict__ b2,
                             const float* __restrict__ scale,
                             const float* __restrict__ shift,
                             float* __restrict__ out) {
  __shared__ float s_scale[HID];
  __shared__ float s_shift[HID];
  for (int i = threadIdx.x; i < HID; i += blockDim.x) {
    s_scale[i] = scale[i];
    s_shift[i] = shift[i];
  }
  __syncthreads();

  int wave = (blockIdx.x * blockDim.x + threadIdx.x) >> 5;
  int lane = threadIdx.x & 31;
  int tileRow = wave >> 3;
  int tileCol = wave & 7;
  int rowBase = tileRow * 16;
  int colBase = tileCol * 16;
  int m = lane & 15;
  int kh = (lane >> 4) << 1;

  v8f c = {};

  // GEMM A: leaky(BN(h1)) @ W2^T
  {
    const float* arow = h1 + (size_t)(rowBase + m) * HID + kh;
    const float* brow = W2 + (size_t)(colBase + m) * HID + kh;
#pragma unroll
    for (int k0 = 0; k0 < HID; k0 += 4) {
      v2f a, b;
      float a0 = arow[k0]     * s_scale[k0 + kh]     + s_shift[k0 + kh];
      float a1 = arow[k0 + 1] * s_scale[k0 + kh + 1] + s_shift[k0 + kh + 1];
      a.x = a0 > 0.f ? a0 : 0.01f * a0;
      a.y = a1 > 0.f ? a1 : 0.01f * a1;
      b.x = brow[k0];
      b.y = brow[k0 + 1];
      c = __builtin_amdgcn_wmma_f32_16x16x4_f32(false, a, false, b, (short)0, c,
                                                false, false);
    }
  }

  // GEMM B (residual): + x @ Wres^T, accumulated into the same tile
  {
    const float* arow = x    + (size_t)(rowBase + m) * IN_DIM + kh;
    const float* brow = Wres + (size_t)(colBase + m) * IN_DIM + kh;
#pragma unroll
    for (int k0 = 0; k0 < IN_DIM; k0 += 4) {
      v2f a, b;
      a.x = arow[k0];  a.y = arow[k0 + 1];
      b.x = brow[k0];  b.y = brow[k0 + 1];
      c = __builtin_amdgcn_wmma_f32_16x16x4_f32(false, a, false, b, (short)0, c,
                                                false, false);
    }
  }

  int col = colBase + m;
  float bias = b2[col];
  int r0 = rowBase + ((lane >> 4) << 3);
#pragma unroll
  for (int i = 0; i < 8; ++i) {
    float v = c[i] + bias;
    out[(size_t)(r0 + i) * HID + col] = v > 0.f ? v : 0.2f * v;
  }
}

// ---------------------------------------------------------------------------
extern "C" void kernel_launch(void* const* d_in, const int* in_sizes, int n_in,
                              void* d_out, int out_size, void* d_ws, size_t ws_size,
                              hipStream_t stream) {
  const float* x       = (const float*)d_in[0];
  const long long* ei  = (const long long*)d_in[1];  // int64 edge_index [2, E]
  const float* W1      = (const float*)d_in[2];
  const float* b1      = (const float*)d_in[3];
  const float* gamma   = (const float*)d_in[4];
  const float* beta    = (const float*)d_in[5];
  const float* W2      = (const float*)d_in[6];
  const float* b2      = (const float*)d_in[7];
  const float* Wres    = (const float*)d_in[8];
  float* out = (float*)d_out;

  float* ws    = (float*)d_ws;
  float* h     = ws;                                   // [N, 128]
  float* h1    = h  + (size_t)N_NODES * HID;           // [N, 128]
  float* sum   = h1 + (size_t)N_NODES * HID;           // [128]
  float* sumsq = sum + HID;                            // [128]
  float* scale = sumsq + HID;                          // [128]
  float* shift = scale + HID;                          // [128]

  init_h_kernel<<<2048, 256, 0, stream>>>(x, h, sum);

  scatter_kernel<<<(N_EDGES * 32) / 256, 256, 0, stream>>>(x, ei, h);

  const int gemmBlocks = (N_NODES / 16) * (HID / 16) / 8;  // 8 waves per block
  gemm1_kernel<<<gemmBlocks, 256, 0, stream>>>(h, W1, b1, h1);

  stats_kernel<<<200, 256, 0, stream>>>(h1, sum, sumsq);
  finalize_kernel<<<1, 128, 0, stream>>>(sum, sumsq, gamma, beta, scale, shift);

  gemm2_kernel<<<gemmBlocks, 256, 0, stream>>>(h1, x, W2, Wres, b2, scale, shift, out);
}